// FrameReducer_11914239279964
// MI455X (gfx1250) — compile-verified
//
#include <hip/hip_runtime.h>
#include <hip/hip_bf16.h>

#define B_ 16
#define T_ 4096
#define C_ 512
#define V_ 512
#define BLANK_THRESH 0.9f

// -----------------------------------------------------------------------------
// Kernel A: per-batch-row stream-compaction scan.
// One 256-thread block per batch row. For each chunk of 256 frames, compute
// keep = (posterior[b,t,blank] < 0.9), inclusive-scan the flags in LDS
// (Hillis-Steele), and record the inverse map src_idx[b, pos] = t for kept
// frames. Also emits kept count (for the gather) and x_lens_fr (float) into
// the tail of d_out.
// -----------------------------------------------------------------------------
__global__ void __launch_bounds__(256) fr_scan_kernel(
    const float* __restrict__ ctc,      // [B,T,V]
    const int*   __restrict__ x_lens,   // [B]
    const int*   __restrict__ blank_p,  // scalar (device)
    int*   __restrict__ src_idx,        // ws: [B,T]
    int*   __restrict__ kept,           // ws: [B]
    float* __restrict__ out_lens)       // d_out tail: [B]
{
    __shared__ int s[256];
    const int b   = blockIdx.x;
    const int tid = threadIdx.x;
    const int blank = *blank_p;
    const float* post = ctc + (size_t)b * T_ * V_ + blank;

    int base = 0;
    for (int c = 0; c < T_ / 256; ++c) {
        const int t = c * 256 + tid;
        const int k = (post[(size_t)t * V_] < BLANK_THRESH) ? 1 : 0;
        s[tid] = k;
        __syncthreads();
        // inclusive scan of 256 flags
        for (int off = 1; off < 256; off <<= 1) {
            const int add = (tid >= off) ? s[tid - off] : 0;
            __syncthreads();
            s[tid] += add;
            __syncthreads();
        }
        const int incl = s[tid];
        if (k) {
            const int pos = base + incl - 1;   // exclusive prefix
            src_idx[b * T_ + pos] = t;
        }
        const int total = s[255];
        __syncthreads();                        // protect s[] before next chunk
        base += total;
    }

    if (tid == 0) {
        kept[b]     = base;                              // frames surviving
        out_lens[b] = (float)(x_lens[b] - (T_ - base));  // x_lens_fr
    }
}

// -----------------------------------------------------------------------------
// Kernel B: gather/compact rows via the CDNA5 async Global<->LDS DMA path.
// One 128-thread block per output row (B*T rows). Each thread moves 16 bytes:
//   global_load_async_to_lds_b128  (HBM -> LDS, ASYNCcnt)
//   s_wait_asynccnt 0
//   global_store_async_from_lds_b128 (LDS -> HBM, ASYNCcnt)
// bypassing the VGPR file for the 2KB row copy. Tail rows (t >= kept) are
// zero-filled directly, so every d_out element is written each call.
// -----------------------------------------------------------------------------
__global__ void __launch_bounds__(128) fr_gather_kernel(
    const float* __restrict__ x,        // [B,T,C]
    const int*   __restrict__ src_idx,  // [B,T]
    const int*   __restrict__ kept,     // [B]
    float* __restrict__ out)            // [B,T,C]
{
    __shared__ float buf[C_];           // 2KB staging, one slice per thread
    const int row = blockIdx.x;
    const int b   = row >> 12;          // row / T_
    const int t   = row & (T_ - 1);
    const int tid = threadIdx.x;
    const size_t outoff = (size_t)row * C_ + (size_t)tid * 4;

    if (t < kept[b]) {
        const int sfrm = src_idx[row];
        const float* gsrc = x + ((size_t)b * T_ + sfrm) * C_ + (size_t)tid * 4;
        float*       gdst = out + outoff;
        // LDS byte offset of this thread's 16B slice (generic addr low 32 bits)
        unsigned lds = (unsigned)(size_t)(&buf[tid * 4]);

        asm volatile("global_load_async_to_lds_b128 %0, %1, off"
                     :: "v"(lds), "v"(gsrc) : "memory");
        asm volatile("s_wait_asynccnt 0x0" ::: "memory");
        asm volatile("global_store_async_from_lds_b128 %0, %1, off"
                     :: "v"(gdst), "v"(lds) : "memory");
        asm volatile("s_wait_asynccnt 0x0" ::: "memory");
    } else {
        const float4 z = make_float4(0.f, 0.f, 0.f, 0.f);
        *(float4*)(out + outoff) = z;
    }
}

// -----------------------------------------------------------------------------
// Launch: inputs are (x, x_lens, ctc_posterior, blank_id) per setup_inputs().
// d_out = x_fr [B*T*C floats] then x_lens_fr [B] (written as float).
// d_ws  = src_idx [B*T ints] + kept [B ints].
// -----------------------------------------------------------------------------
extern "C" void kernel_launch(void* const* d_in, const int* in_sizes, int n_in,
                              void* d_out, int out_size, void* d_ws, size_t ws_size,
                              hipStream_t stream) {
    const float* x       = (const float*)d_in[0];
    const int*   x_lens  = (const int*)  d_in[1];
    const float* ctc     = (const float*)d_in[2];
    const int*   blank_p = (const int*)  d_in[3];

    float* out_x    = (float*)d_out;
    float* out_lens = out_x + (size_t)B_ * T_ * C_;

    int* src_idx = (int*)d_ws;
    int* kept    = src_idx + (size_t)B_ * T_;

    // Pass 1: keep-flag scan -> inverse compaction map + lengths
    fr_scan_kernel<<<B_, 256, 0, stream>>>(ctc, x_lens, blank_p,
                                           src_idx, kept, out_lens);

    // Pass 2: row gather through async Global<->LDS DMA (+ zero tail)
    fr_gather_kernel<<<B_ * T_, 128, 0, stream>>>(x, src_idx, kept, out_x);
}